// TourLayer_35708358099144
// MI455X (gfx1250) — compile-verified
//
#include <hip/hip_runtime.h>
#include <math.h>

typedef __attribute__((ext_vector_type(2))) float v2f;
typedef __attribute__((ext_vector_type(8))) float v8f;

#define B_    64
#define P_    2000
#define D_    128
#define T_    200
#define ASTR  260   // 16B-aligned, 260%64==4 -> conflict-free strided rows
#define HSTR  132   // 132%64==4

__global__ __launch_bounds__(256) void zero_kernel(float* __restrict__ p, int n) {
  int i = blockIdx.x * 256 + threadIdx.x;
  if (i < n) p[i] = 0.f;
}

// Pack row-major W (K x 128) into WMMA B-fragment order:
// Wp[nt][ks][lane] = { W[4*ks+2*lh][nt*16+lm], W[4*ks+2*lh+1][nt*16+lm] }
// so the inner loop fetches one coalesced global_load_b64 per fragment.
__global__ __launch_bounds__(256) void pack_kernel(const float* __restrict__ W,
                                                   float2* __restrict__ Wp, int K) {
  int ksteps = K >> 2;
  int total = 8 * ksteps * 32;
  int i = blockIdx.x * 256 + threadIdx.x;
  if (i >= total) return;
  int lane = i & 31;
  int ks   = (i >> 5) % ksteps;
  int nt   = i / (ksteps * 32);
  int lm = lane & 15, lh = lane >> 4;
  int ka = 4 * ks + 2 * lh;
  float2 v;
  v.x = W[(size_t)ka * D_ + nt * 16 + lm];
  v.y = W[(size_t)(ka + 1) * D_ + nt * 16 + lm];
  Wp[i] = v;
}

// One block per (batch b, 64-wide d-slice). 200x64 fp32 accumulators in LDS.
__global__ __launch_bounds__(256) void segsum_kernel(const float* __restrict__ emb,
                                                     const int* __restrict__ tidx,
                                                     float* __restrict__ tour) {
  __shared__ float acc[T_ * 64];
  int b  = blockIdx.x >> 1;
  int d0 = (blockIdx.x & 1) << 6;
  int tid = threadIdx.x;
  for (int i = tid; i < T_ * 64; i += 256) acc[i] = 0.f;
  __syncthreads();
  int d  = tid & 63;
  int ps = tid >> 6;          // 4 customer rows in flight
  for (int p = ps; p < P_; p += 4) {
    int t = tidx[b * P_ + p];
    float v = emb[((size_t)(b * (P_ + 1) + 1 + p)) * D_ + d0 + d];
    atomicAdd(&acc[t * 64 + d], v);   // ds_add_f32
  }
  __syncthreads();
  for (int i = tid; i < T_ * 64; i += 256) {
    int t = i >> 6, dd = i & 63;
    tour[((size_t)(b * T_ + t)) * D_ + d0 + dd] = acc[i];
  }
}

// One block (4 waves) per 16-row tile: fused concat + GEMM1 + ReLU + GEMM2 +
// residual, writes `added` into d_out rows 1..P, accumulates column sum/sumsq.
__global__ __launch_bounds__(128) void mlp_kernel(
    const float* __restrict__ emb,   const int* __restrict__ tidx,
    const float2* __restrict__ Wp1,  const float* __restrict__ b1,
    const float2* __restrict__ Wp2,  const float* __restrict__ b2,
    const float* __restrict__ tour,  float* __restrict__ out,
    float* __restrict__ s1,          float* __restrict__ s2)
{
  __shared__ float As[16 * ASTR];   // [cust | cust_tour] 16 x 256
  __shared__ float Hs[16 * HSTR];   // relu(h) 16 x 128

  int bid = blockIdx.x;
  int b   = bid / (P_ / 16);
  int p0  = (bid % (P_ / 16)) * 16;
  int tid = threadIdx.x;
  int w    = tid >> 5;
  int lane = tid & 31;
  int lm   = lane & 15;     // M (A) / N (B,C)
  int lh   = lane >> 4;     // K-half (A,B) / M+8 (C)

  // Stage A: cust tile then gathered tour tile (float4 global + LDS)
  for (int i = tid; i < 16 * 32; i += 128) {
    int M = i >> 5, c = (i & 31) << 2;
    float4 v = *(const float4*)&emb[((size_t)(b * (P_ + 1) + 1 + p0 + M)) * D_ + c];
    *(float4*)&As[M * ASTR + c] = v;
  }
  for (int i = tid; i < 16 * 32; i += 128) {
    int M = i >> 5, c = (i & 31) << 2;
    int t = tidx[b * P_ + p0 + M];
    float4 v = *(const float4*)&tour[((size_t)(b * T_ + t)) * D_ + c];
    *(float4*)&As[M * ASTR + D_ + c] = v;
  }
  __syncthreads();

  int n0 = w * 16, n1 = (w + 4) * 16;
  const float2* Wp1a = Wp1 + ((size_t)w * 64) * 32 + lane;        // ntile w
  const float2* Wp1b = Wp1 + ((size_t)(w + 4) * 64) * 32 + lane;  // ntile w+4
  const float2* Wp2a = Wp2 + ((size_t)w * 32) * 32 + lane;
  const float2* Wp2b = Wp2 + ((size_t)(w + 4) * 32) * 32 + lane;

  // GEMM1: (16x256) @ W1(256x128), fp32 WMMA, K-step 4, packed-b64 B frags
  v8f acc0 = {}; v8f acc1 = {};
#pragma unroll 8
  for (int ks = 0; ks < 64; ++ks) {
    int ka = 4 * ks + 2 * lh;
    v2f a; a.x = As[lm * ASTR + ka]; a.y = As[lm * ASTR + ka + 1];
    float2 q0 = Wp1a[ks * 32];
    float2 q1 = Wp1b[ks * 32];
    v2f bf0; bf0.x = q0.x; bf0.y = q0.y;
    v2f bf1; bf1.x = q1.x; bf1.y = q1.y;
    acc0 = __builtin_amdgcn_wmma_f32_16x16x4_f32(false, a, false, bf0, (short)0, acc0, false, false);
    acc1 = __builtin_amdgcn_wmma_f32_16x16x4_f32(false, a, false, bf1, (short)0, acc1, false, false);
  }
  float bias0 = b1[n0 + lm], bias1 = b1[n1 + lm];
#pragma unroll
  for (int r = 0; r < 8; ++r) {
    int M = r + 8 * lh;
    float h0 = acc0[r] + bias0; h0 = h0 > 0.f ? h0 : 0.f;
    float h1 = acc1[r] + bias1; h1 = h1 > 0.f ? h1 : 0.f;
    Hs[M * HSTR + n0 + lm] = h0;
    Hs[M * HSTR + n1 + lm] = h1;
  }
  __syncthreads();

  // GEMM2: (16x128) @ W2(128x128)
  v8f d0 = {}; v8f d1 = {};
#pragma unroll 8
  for (int ks = 0; ks < 32; ++ks) {
    int ka = 4 * ks + 2 * lh;
    v2f a; a.x = Hs[lm * HSTR + ka]; a.y = Hs[lm * HSTR + ka + 1];
    float2 q0 = Wp2a[ks * 32];
    float2 q1 = Wp2b[ks * 32];
    v2f bf0; bf0.x = q0.x; bf0.y = q0.y;
    v2f bf1; bf1.x = q1.x; bf1.y = q1.y;
    d0 = __builtin_amdgcn_wmma_f32_16x16x4_f32(false, a, false, bf0, (short)0, d0, false, false);
    d1 = __builtin_amdgcn_wmma_f32_16x16x4_f32(false, a, false, bf1, (short)0, d1, false, false);
  }
  float c0 = b2[n0 + lm], c1 = b2[n1 + lm];
  float sum0 = 0.f, sq0 = 0.f, sum1 = 0.f, sq1 = 0.f;
#pragma unroll
  for (int r = 0; r < 8; ++r) {
    int M = r + 8 * lh;
    size_t row = (size_t)(b * (P_ + 1) + 1 + p0 + M) * D_;
    float a0 = As[M * ASTR + n0 + lm] + d0[r] + c0;  // cust + h
    float a1 = As[M * ASTR + n1 + lm] + d1[r] + c1;
    out[row + n0 + lm] = a0;
    out[row + n1 + lm] = a1;
    sum0 += a0; sq0 += a0 * a0;
    sum1 += a1; sq1 += a1 * a1;
  }
  atomicAdd(&s1[b * D_ + n0 + lm], sum0);
  atomicAdd(&s2[b * D_ + n0 + lm], sq0);
  atomicAdd(&s1[b * D_ + n1 + lm], sum1);
  atomicAdd(&s2[b * D_ + n1 + lm], sq1);
}

// Normalize in place over the customer axis; copy depot row.
__global__ __launch_bounds__(128) void norm_kernel(
    const float* __restrict__ emb, const float* __restrict__ s1,
    const float* __restrict__ s2,  const float* __restrict__ gamma,
    const float* __restrict__ beta, float* __restrict__ out)
{
  int row = blockIdx.x;
  int b = row / (P_ + 1);
  int p = row % (P_ + 1);
  int d = threadIdx.x;
  size_t o = (size_t)row * D_ + d;
  if (p == 0) { out[o] = emb[o]; return; }
  const float inv = 1.0f / (float)P_;
  float mean = s1[b * D_ + d] * inv;
  float m2   = s2[b * D_ + d] * inv;
  float var  = m2 - mean * mean;
  float x = out[o];
  out[o] = (x - mean) * rsqrtf(var + 1e-5f) * gamma[d] + beta[d];
}

extern "C" void kernel_launch(void* const* d_in, const int* in_sizes, int n_in,
                              void* d_out, int out_size, void* d_ws, size_t ws_size,
                              hipStream_t stream) {
  (void)in_sizes; (void)n_in; (void)out_size; (void)ws_size;
  const int*   tidx  = (const int*)d_in[2];
  const float* emb   = (const float*)d_in[3];
  const float* W1    = (const float*)d_in[4];
  const float* b1    = (const float*)d_in[5];
  const float* W2    = (const float*)d_in[6];
  const float* b2    = (const float*)d_in[7];
  const float* gamma = (const float*)d_in[8];
  const float* beta  = (const float*)d_in[9];
  float* out = (float*)d_out;

  char* ws = (char*)d_ws;
  size_t off = 0;
  float*  tour = (float*)(ws + off);  off += (size_t)B_ * T_ * D_ * 4;   // 6.55 MB
  float*  s1   = (float*)(ws + off);  off += (size_t)B_ * D_ * 4;
  float*  s2   = (float*)(ws + off);  off += (size_t)B_ * D_ * 4;
  float2* Wp1  = (float2*)(ws + off); off += (size_t)2 * D_ * D_ * 4;    // 128 KB
  float2* Wp2  = (float2*)(ws + off); off += (size_t)D_ * D_ * 4;        // 64 KB

  zero_kernel<<<(2 * B_ * D_ + 255) / 256, 256, 0, stream>>>(s1, 2 * B_ * D_);
  pack_kernel<<<(8 * 64 * 32 + 255) / 256, 256, 0, stream>>>(W1, Wp1, 2 * D_);
  pack_kernel<<<(8 * 32 * 32 + 255) / 256, 256, 0, stream>>>(W2, Wp2, D_);
  segsum_kernel<<<B_ * 2, 256, 0, stream>>>(emb, tidx, tour);
  mlp_kernel<<<B_ * (P_ / 16), 128, 0, stream>>>(emb, tidx, Wp1, b1, Wp2, b2, tour, out, s1, s2);
  norm_kernel<<<B_ * (P_ + 1), 128, 0, stream>>>(emb, s1, s2, gamma, beta, out);
}